// CylindricalEnergyMLP_52501680227090
// MI455X (gfx1250) — compile-verified
//
#include <hip/hip_runtime.h>
#include <hip/hip_bf16.h>
#include <math.h>

// ---------------- problem constants ----------------
#define OBS_C   128
#define MLPC    256
#define LMAXC   3
#define RFC     4
#define AFC     4
#define AANGC   3
#define NGRIDC  16
#define NRC     64
#define NPC     128
#define NHC     64
#define GC      64           // B * NACT = 2 * 32
#define MC      7            // 2*AANG+1
#define KQ      28           // AF * M
#define NPH     8192         // NP * NH
#define K1PAD   132          // 129 padded to multiple of 4

typedef __attribute__((ext_vector_type(2))) float v2f;
typedef __attribute__((ext_vector_type(8))) float v8f;

#ifndef M_PIF
#define M_PIF 3.14159265358979323846f
#endif

// =====================================================================
// Generic batched WMMA f32 GEMM:  C[b] = A[b](MxK) * B[b](KxN) (+bias)
//   - A row-major [row][k], lda even, KT multiple of 4 (compile-time)
//   - Bm stored as [n][k] (weight layout), ldb even
//   - One wave per PAIR of adjacent 16x16 C tiles (shared A fragment,
//     two independent WMMA accumulation chains).
//   - KT compile-time: K=28 (final expansion) fully unrolls.
//   - NT: non-temporal stores for the streaming 134MB output.
// =====================================================================
template <int KT, bool NT>
__global__ void gemm_wmma_f32(const float* __restrict__ A,
                              const float* __restrict__ Bm,
                              const float* __restrict__ bias,
                              float* __restrict__ C,
                              int M, int N,
                              int lda, int ldb, int ldc,
                              long sA, long sB, long sC,
                              int nbatch) {
    const int ntm  = M >> 4;
    const int ntn  = N >> 4;
    const int ntn2 = (ntn + 1) >> 1;              // tile pairs along N
    const int tilesPerBatch = ntm * ntn2;
    const int wave = blockIdx.x * (blockDim.x >> 5) + (threadIdx.x >> 5);
    const int batch = wave / tilesPerBatch;
    if (batch >= nbatch) return;                  // whole-wave exit, EXEC stays full
    const int tile = wave - batch * tilesPerBatch;
    const int tm  = tile / ntn2;
    const int tn0 = (tile - tm * ntn2) * 2;
    const bool has2 = (tn0 + 1) < ntn;

    const int lane  = threadIdx.x & 31;
    const int half  = lane >> 4;      // 0 or 1
    const int idx16 = lane & 15;
    const int kb    = half << 1;      // 0 or 2 : K offset within a 4-step
    const int arow  = tm * 16 + idx16;
    const int bcol0 = tn0 * 16 + idx16;
    const int bcol1 = has2 ? (bcol0 + 16) : bcol0;   // clamp: duplicate reads if no 2nd tile

    // per-lane base pointers (8-byte aligned: lda/ldb even, kb even)
    const float* __restrict__ ap  = A  + (long)batch * sA + (long)arow  * lda + kb;
    const float* __restrict__ bp0 = Bm + (long)batch * sB + (long)bcol0 * ldb + kb;
    const float* __restrict__ bp1 = Bm + (long)batch * sB + (long)bcol1 * ldb + kb;

    v8f acc0 = {};
    v8f acc1 = {};
#pragma unroll
    for (int k0 = 0; k0 < KT; k0 += 4) {
        v2f av  = *(const v2f*)(ap  + k0);
        v2f bv0 = *(const v2f*)(bp0 + k0);
        v2f bv1 = *(const v2f*)(bp1 + k0);
        acc0 = __builtin_amdgcn_wmma_f32_16x16x4_f32(
                   false, av, false, bv0, (short)0, acc0, false, false);
        acc1 = __builtin_amdgcn_wmma_f32_16x16x4_f32(
                   false, av, false, bv1, (short)0, acc1, false, false);
    }

    float* __restrict__ Cb = C + (long)batch * sC;
    const int ccol0 = tn0 * 16 + idx16;
    float badd0 = bias ? bias[ccol0] : 0.0f;
#pragma unroll
    for (int v = 0; v < 8; ++v) {
        const int crow = tm * 16 + v + half * 8;
        float* p = Cb + (long)crow * ldc + ccol0;
        float val = acc0[v] + badd0;
        if (NT) __builtin_nontemporal_store(val, p);
        else    *p = val;
    }
    if (has2) {
        const int ccol1 = ccol0 + 16;
        float badd1 = bias ? bias[ccol1] : 0.0f;
#pragma unroll
        for (int v = 0; v < 8; ++v) {
            const int crow = tm * 16 + v + half * 8;
            float* p = Cb + (long)crow * ldc + ccol1;
            float val = acc1[v] + badd1;
            if (NT) __builtin_nontemporal_store(val, p);
            else    *p = val;
        }
    }
}

// =====================================================================
// Input assembly (X0 padded to 132 cols, zeros in the tail)
// =====================================================================
__global__ void assemble_x0(const float* __restrict__ obs,
                            const float* __restrict__ act,
                            float* __restrict__ X0) {
    int idx = blockIdx.x * blockDim.x + threadIdx.x;      // 64*132
    if (idx >= GC * K1PAD) return;
    int g = idx / K1PAD;
    int c = idx - g * K1PAD;
    int b = g >> 5;                                       // NACT = 32
    float v = 0.0f;
    if (c < OBS_C)       v = obs[b * (OBS_C * MC) + c * MC];
    else if (c == OBS_C) v = act[g];
    X0[idx] = v;
}

__global__ void assemble_xl(const float* __restrict__ obs,
                            float* __restrict__ XL) {     // [3][64][256]: [xc|xs]
    int idx = blockIdx.x * blockDim.x + threadIdx.x;      // 3*64*128
    if (idx >= LMAXC * GC * OBS_C) return;
    int l = idx / (GC * OBS_C);
    int r = idx - l * (GC * OBS_C);
    int g = r / OBS_C;
    int i = r - g * OBS_C;
    int b = g >> 5;
    const float* row = obs + b * (OBS_C * MC) + i * MC;
    float* dst = XL + (long)l * GC * 256 + g * 256;
    dst[i]         = row[1 + 2 * l];   // xc
    dst[OBS_C + i] = row[2 + 2 * l];   // xs
}

// W0_1 (256x129) -> zero-padded (256x132)
__global__ void pack_w01(const float* __restrict__ W01,
                         float* __restrict__ W01P) {
    int idx = blockIdx.x * blockDim.x + threadIdx.x;      // 256*132
    if (idx >= MLPC * K1PAD) return;
    int o = idx / K1PAD;
    int k = idx - o * K1PAD;
    W01P[idx] = (k < OBS_C + 1) ? W01[o * (OBS_C + 1) + k] : 0.0f;
}

// =====================================================================
// SO(2) block-weight packing:  W = [[a, -b], [b, a]]   (rows = outputs)
// =====================================================================
__global__ void pack_wl(const float* __restrict__ AB,  // [l][2][NO][NI]
                        float* __restrict__ W,         // [l][2*NO][2*NI]
                        int nl, int NO, int NI) {
    int idx = blockIdx.x * blockDim.x + threadIdx.x;
    int per = 2 * NO * 2 * NI;
    if (idx >= nl * per) return;
    int l = idx / per;
    int r = idx - l * per;
    int o = r / (2 * NI);
    int k = r - o * (2 * NI);
    const float* a = AB + ((long)l * 2 + 0) * NO * NI;
    const float* b = AB + ((long)l * 2 + 1) * NO * NI;
    float v;
    if (o < NO) {
        v = (k < NI) ? a[(long)o * NI + k] : -b[(long)o * NI + (k - NI)];
    } else {
        int oo = o - NO;
        v = (k < NI) ? b[(long)oo * NI + k] : a[(long)oo * NI + (k - NI)];
    }
    W[idx] = v;
}

// =====================================================================
// Fourier ELU nonlinearity (per (g, channel))
// =====================================================================
__global__ void fourier_elu(const float* __restrict__ Y0,
                            const float* __restrict__ YL,
                            float* __restrict__ X0,
                            float* __restrict__ XL) {
    int idx = blockIdx.x * blockDim.x + threadIdx.x;      // 64*256
    if (idx >= GC * MLPC) return;
    int g = idx >> 8;
    int o = idx & 255;
    float y0 = Y0[idx];
    float yc[LMAXC], ys[LMAXC];
#pragma unroll
    for (int l = 0; l < LMAXC; ++l) {
        const float* row = YL + (long)l * GC * 512 + g * 512;
        yc[l] = row[o];
        ys[l] = row[MLPC + o];
    }
    float z0 = 0.0f, zc[LMAXC] = {0.f, 0.f, 0.f}, zs[LMAXC] = {0.f, 0.f, 0.f};
#pragma unroll
    for (int j = 0; j < NGRIDC; ++j) {
        float th = (float)j * (2.0f * M_PIF / (float)NGRIDC);
        float cj[LMAXC], sj[LMAXC];
        float v = y0;
#pragma unroll
        for (int l = 0; l < LMAXC; ++l) {
            float a = (float)(l + 1) * th;
            cj[l] = __cosf(a);
            sj[l] = __sinf(a);
            v += yc[l] * cj[l] + ys[l] * sj[l];
        }
        float w = (v > 0.0f) ? v : (__expf(v) - 1.0f);
        z0 += w;
#pragma unroll
        for (int l = 0; l < LMAXC; ++l) {
            zc[l] += w * cj[l];
            zs[l] += w * sj[l];
        }
    }
    X0[idx] = z0 * (1.0f / (float)NGRIDC);
#pragma unroll
    for (int l = 0; l < LMAXC; ++l) {
        float* row = XL + (long)l * GC * 512 + g * 512;
        row[o]        = zc[l] * (2.0f / (float)NGRIDC);
        row[MLPC + o] = zs[l] * (2.0f / (float)NGRIDC);
    }
}

// =====================================================================
// Q[g][r][kq] = c1[g,k,m,r] = sum_n coeffs[g, n*4+k, m] * rad[m][n][r]
// =====================================================================
__global__ void build_q(const float* __restrict__ C0,   // [64][16]  (y0)
                        const float* __restrict__ CL,   // [3][64][32] ([yc|ys])
                        const float* __restrict__ rad,  // [7][4][64]
                        float* __restrict__ Q) {
    int idx = blockIdx.x * blockDim.x + threadIdx.x;     // 64*64*28
    if (idx >= GC * NRC * KQ) return;
    int g  = idx / (NRC * KQ);
    int r2 = idx - g * (NRC * KQ);
    int r  = r2 / KQ;
    int kq = r2 - r * KQ;
    int k = kq / MC;
    int m = kq - k * MC;
    float sum = 0.0f;
#pragma unroll
    for (int n = 0; n < RFC; ++n) {
        int o = n * AFC + k;
        float cf;
        if (m == 0) {
            cf = C0[g * 16 + o];
        } else {
            int l = (m - 1) >> 1;
            int t = (m - 1) & 1;
            cf = CL[(long)l * GC * 32 + g * 32 + t * 16 + o];
        }
        sum += cf * rad[(m * RFC + n) * NRC + r];
    }
    Q[idx] = sum;
}

// =====================================================================
// ET[col][kq] = ang[m][p] * ax[k][h]   (transposed: weight [n][k] layout)
//   col = p*64+h, kq = k*7+m
// =====================================================================
__global__ void build_et(float* __restrict__ ET) {
    int idx = blockIdx.x * blockDim.x + threadIdx.x;     // 8192*28
    if (idx >= NPH * KQ) return;
    int col = idx / KQ;
    int kq  = idx - col * KQ;
    int k = kq / MC;
    int m = kq - k * MC;
    int p = col >> 6;
    int h = col & 63;
    float ang;
    if (m == 0) {
        ang = 0.3989422804014327f;                 // 1/sqrt(2*pi)
    } else {
        int   mi  = (m + 1) >> 1;
        float phi = (float)p * (2.0f * M_PIF / (float)NPC);
        float a   = (float)mi * phi;
        float t   = ((m & 1) == 1) ? __cosf(a) : __sinf(a);
        ang = t * 0.5641895835477563f;             // 1/sqrt(pi)
    }
    float z  = (float)h * (1.0f / (float)(NHC - 1));
    float ax = 1.4142135623730951f * __sinf((float)(k + 1) * M_PIF * z);
    ET[idx] = ang * ax;
}

// =====================================================================
// host-side launcher
// =====================================================================
extern "C" void kernel_launch(void* const* d_in, const int* in_sizes, int n_in,
                              void* d_out, int out_size, void* d_ws, size_t ws_size,
                              hipStream_t stream) {
    (void)in_sizes; (void)n_in; (void)out_size; (void)ws_size;
    const float* obs   = (const float*)d_in[0];   // (2, 896)
    const float* act   = (const float*)d_in[1];   // (64,)
    const float* W01   = (const float*)d_in[2];   // (256,129)
    const float* b01   = (const float*)d_in[3];   // (256,)
    const float* AB1   = (const float*)d_in[4];   // (3,2,256,128)
    const float* W0h   = (const float*)d_in[5];   // (2,256,256)
    const float* b0h   = (const float*)d_in[6];   // (2,256)
    const float* ABh   = (const float*)d_in[7];   // (2,3,2,256,256)
    const float* Wout0 = (const float*)d_in[8];   // (16,256)
    const float* bout0 = (const float*)d_in[9];   // (16,)
    const float* ABout = (const float*)d_in[10];  // (3,2,16,256)
    const float* rad   = (const float*)d_in[11];  // (7,4,64)
    float* out = (float*)d_out;
    float* ws  = (float*)d_ws;

    // workspace layout (floats)
    size_t off = 0;
    float* XIN0  = ws + off; off += (size_t)GC * K1PAD;       // layer-1 scalar input (padded)
    float* W01P  = ws + off; off += (size_t)MLPC * K1PAD;     // padded W0_1
    float* XINL  = ws + off; off += (size_t)LMAXC * GC * 256; // layer-1 [xc|xs]
    float* X0A   = ws + off; off += (size_t)GC * MLPC;
    float* X0B   = ws + off; off += (size_t)GC * MLPC;
    float* XLA   = ws + off; off += (size_t)LMAXC * GC * 512;
    float* XLB   = ws + off; off += (size_t)LMAXC * GC * 512;
    float* Y0    = ws + off; off += (size_t)GC * MLPC;
    float* YL    = ws + off; off += (size_t)LMAXC * GC * 512;
    float* WL1   = ws + off; off += (size_t)LMAXC * 512 * 256;
    float* WLH   = ws + off; off += (size_t)2 * LMAXC * 512 * 512;
    float* WOUTL = ws + off; off += (size_t)LMAXC * 32 * 512;
    float* C0O   = ws + off; off += (size_t)GC * 16;
    float* CLO   = ws + off; off += (size_t)LMAXC * GC * 32;
    float* Q     = ws + off; off += (size_t)GC * NRC * KQ;
    float* ET    = ws + off; off += (size_t)NPH * KQ;

    #define LAUNCH_GEMM(KT, NT, A_, B_, BIAS_, C_, M_, N_, lda_, ldb_, ldc_, sA_, sB_, sC_, nb_) \
        do {                                                                                      \
            int ntn2_  = (((N_) / 16) + 1) / 2;                                                   \
            int tiles_ = ((M_) / 16) * ntn2_ * (nb_);                                             \
            int blocks_ = (tiles_ + 3) / 4;  /* 4 waves / block */                                \
            gemm_wmma_f32<KT, NT><<<blocks_, 128, 0, stream>>>(                                   \
                A_, B_, BIAS_, C_, M_, N_, lda_, ldb_, ldc_,                                      \
                (long)(sA_), (long)(sB_), (long)(sC_), nb_);                                      \
        } while (0)

    // ---- constant / weight preparation (deterministic, every call) ----
    {
        int n;
        n = GC * K1PAD;
        assemble_x0<<<(n + 255) / 256, 256, 0, stream>>>(obs, act, XIN0);
        n = MLPC * K1PAD;
        pack_w01<<<(n + 255) / 256, 256, 0, stream>>>(W01, W01P);
        n = LMAXC * GC * OBS_C;
        assemble_xl<<<(n + 255) / 256, 256, 0, stream>>>(obs, XINL);
        n = LMAXC * 512 * 256;
        pack_wl<<<(n + 255) / 256, 256, 0, stream>>>(AB1, WL1, LMAXC, MLPC, OBS_C);
        n = 2 * LMAXC * 512 * 512;  // ABh viewed as (2*3) blocks of [2][256][256]
        pack_wl<<<(n + 255) / 256, 256, 0, stream>>>(ABh, WLH, 2 * LMAXC, MLPC, MLPC);
        n = LMAXC * 32 * 512;
        pack_wl<<<(n + 255) / 256, 256, 0, stream>>>(ABout, WOUTL, LMAXC, 16, MLPC);
        n = NPH * KQ;
        build_et<<<(n + 255) / 256, 256, 0, stream>>>(ET);
    }

    // ---- layer 1 ----
    LAUNCH_GEMM(K1PAD, false, XIN0, W01P, b01, Y0, GC, MLPC,
                K1PAD, K1PAD, MLPC, 0, 0, 0, 1);
    LAUNCH_GEMM(256, false, XINL, WL1, nullptr, YL, GC, 512,
                256, 256, 512, (long)GC * 256, (long)512 * 256, (long)GC * 512, LMAXC);
    fourier_elu<<<(GC * MLPC + 255) / 256, 256, 0, stream>>>(Y0, YL, X0A, XLA);

    // ---- hidden layers ----
    float* x0cur = X0A; float* xlcur = XLA;
    float* x0nxt = X0B; float* xlnxt = XLB;
    for (int i = 0; i < 2; ++i) {
        LAUNCH_GEMM(256, false, x0cur, W0h + (size_t)i * MLPC * MLPC,
                    b0h + (size_t)i * MLPC, Y0, GC, MLPC,
                    MLPC, MLPC, MLPC, 0, 0, 0, 1);
        LAUNCH_GEMM(512, false, xlcur, WLH + (size_t)i * LMAXC * 512 * 512,
                    nullptr, YL, GC, 512,
                    512, 512, 512, (long)GC * 512, (long)512 * 512, (long)GC * 512, LMAXC);
        fourier_elu<<<(GC * MLPC + 255) / 256, 256, 0, stream>>>(Y0, YL, x0nxt, xlnxt);
        float* t0 = x0cur; x0cur = x0nxt; x0nxt = t0;
        float* t1 = xlcur; xlcur = xlnxt; xlnxt = t1;
    }

    // ---- output SO(2) layer ----
    LAUNCH_GEMM(256, false, x0cur, Wout0, bout0, C0O, GC, 16,
                MLPC, MLPC, 16, 0, 0, 0, 1);
    LAUNCH_GEMM(512, false, xlcur, WOUTL, nullptr, CLO, GC, 32,
                512, 512, 32, (long)GC * 512, (long)32 * 512, (long)GC * 32, LMAXC);

    // ---- radial contraction -> Q ----
    {
        int n = GC * NRC * KQ;
        build_q<<<(n + 255) / 256, 256, 0, stream>>>(C0O, CLO, rad, Q);
    }

    // ---- final fused expansion: out[g] (64 x 8192) = Q[g] (64x28) * ET^T ----
    // K=28 fully unrolled, non-temporal streaming stores (134 MB, write-once)
    LAUNCH_GEMM(KQ, true, Q, ET, nullptr, out, NRC, NPH,
                KQ, KQ, NPH, (long)NRC * KQ, 0, (long)NRC * NPH, GC);

    #undef LAUNCH_GEMM
}